// GenericHOIDetector_69260642615865
// MI455X (gfx1250) — compile-verified
//
#include <hip/hip_runtime.h>
#include <math.h>

#define QN 900
#define C1N 81
#define NCLS 80
#define DN 256
#define VN 117
#define NBN 8
#define NTOT 125
#define NPAD 128          // padded N for packed weights
#define MAXH 15
#define MAXO 15
#define NPAIR 225
#define NEGV -1000000000.0f
#define SCORE_THF 0.2f
#define NMS_THF 0.5f
#define SORTN 1024
#define NMSW 29           // ceil(900/32) words per suppression row

typedef float v2f __attribute__((ext_vector_type(2)));
typedef float v8f __attribute__((ext_vector_type(8)));

// ---- prep kernel: pack [Wv | Wb | 0] into Wc[512][128] and biasc[128] ----
__global__ __launch_bounds__(256) void prep_w_kernel(
    const float* __restrict__ Wv, const float* __restrict__ bv,
    const float* __restrict__ Wb, const float* __restrict__ bb,
    float* __restrict__ Wc, float* __restrict__ biasc)
{
  int i = blockIdx.x * 256 + threadIdx.x;
  if (i < 2 * DN * NPAD) {
    int k = i >> 7, n = i & (NPAD - 1);
    float v = 0.f;
    if (n < VN)        v = Wv[k * VN + n];
    else if (n < NTOT) v = Wb[k * NBN + (n - VN)];
    Wc[i] = v;
  }
  if (i < NPAD) {
    float v = 0.f;
    if (i < VN)        v = bv[i];
    else if (i < NTOT) v = bb[i - VN];
    biasc[i] = v;
  }
}

__global__ __launch_bounds__(256) void hoi_kernel(
    const float* __restrict__ logits,   // B,900,81
    const float* __restrict__ boxes,    // B,900,4
    const float* __restrict__ hs,       // B,900,256
    const float* __restrict__ sizes,    // B,2
    const float* __restrict__ Wc,       // 512,128 packed
    const float* __restrict__ biasc,    // 128 packed
    float* __restrict__ out_boxes,      // B,30,4
    float* __restrict__ out_bh,         // B,225,4
    float* __restrict__ out_bo,         // B,225,4
    float* __restrict__ out_det)        // B,225,117
{
  __shared__ float s_score[QN];
  __shared__ int   s_label[QN];
  __shared__ float s_key[SORTN];
  __shared__ int   s_idx[SORTN];
  __shared__ float s_box4[QN][4];     // offset boxes in sorted order
  __shared__ int   s_keepO[QN];       // keep (original order)
  __shared__ unsigned s_keepW[NMSW];  // keep bitset (sorted order)
  __shared__ float s_red[256];
  __shared__ float s_bmax;
  __shared__ int   s_idxh[MAXH + 1], s_idxo[MAXO];  // +1 pad slot for GEMM rows >= 225
  __shared__ float s_sch[MAXH],  s_sco[MAXO];
  __shared__ int   s_valh[MAXH], s_valo[MAXO];
  __shared__ float s_pr[NPAIR];
  __shared__ float s_delta[NPAIR][NBN];
  extern __shared__ unsigned s_mask[];  // QN * NMSW suppression bitmask words

  const int b   = blockIdx.x;
  const int tid = threadIdx.x;
  const float* lg = logits + (size_t)b * QN * C1N;
  const float* bx = boxes  + (size_t)b * QN * 4;
  const float* hb = hs     + (size_t)b * QN * DN;

  // ---- Phase 1: softmax scores + labels ----
  for (int q = tid; q < QN; q += 256) {
    const float* row = lg + q * C1N;
    float m = row[0];
    #pragma unroll 1
    for (int c = 1; c < C1N; ++c) m = fmaxf(m, row[c]);
    float mf = row[0]; int lab = 0;
    #pragma unroll 1
    for (int c = 1; c < NCLS; ++c) { float v = row[c]; if (v > mf) { mf = v; lab = c; } }
    float den = 0.f;
    #pragma unroll 1
    for (int c = 0; c < C1N; ++c) den += expf(row[c] - m);
    s_score[q] = expf(mf - m) / den;
    s_label[q] = lab;
  }

  // ---- boxes.max() reduction ----
  {
    float lm = -1e30f;
    for (int i = tid; i < QN * 4; i += 256) lm = fmaxf(lm, bx[i]);
    s_red[tid] = lm;
    __syncthreads();
    for (int s = 128; s > 0; s >>= 1) {
      if (tid < s) s_red[tid] = fmaxf(s_red[tid], s_red[tid + s]);
      __syncthreads();
    }
    if (tid == 0) s_bmax = s_red[0];
    __syncthreads();
  }

  // ---- Phase 2: bitonic argsort (score desc, index asc == stable argsort(-s)) ----
  for (int i = tid; i < SORTN; i += 256) {
    s_key[i] = (i < QN) ? s_score[i] : -1e30f;
    s_idx[i] = i;
  }
  __syncthreads();
  for (int k = 2; k <= SORTN; k <<= 1) {
    for (int j = k >> 1; j > 0; j >>= 1) {
      for (int i = tid; i < SORTN; i += 256) {
        int ixj = i ^ j;
        if (ixj > i) {
          float a = s_key[i], c = s_key[ixj];
          int   ai = s_idx[i], ci = s_idx[ixj];
          bool before = (a > c) || (a == c && ai < ci); // i before ixj in desc order
          bool doswap = ((i & k) == 0) ? (!before) : before;
          if (doswap) {
            s_key[i] = c; s_key[ixj] = a;
            s_idx[i] = ci; s_idx[ixj] = ai;
          }
        }
      }
      __syncthreads();
    }
  }

  // ---- Phase 3: sorted offset boxes (cxcywh treated as xyxy, per reference) ----
  {
    float off_mul = s_bmax + 1.0f;
    for (int r = tid; r < QN; r += 256) {
      int q = s_idx[r];
      float off = (float)s_label[q] * off_mul;
      s_box4[r][0] = bx[q * 4 + 0] + off;
      s_box4[r][1] = bx[q * 4 + 1] + off;
      s_box4[r][2] = bx[q * 4 + 2] + off;
      s_box4[r][3] = bx[q * 4 + 3] + off;
    }
  }
  if (tid < NMSW) s_keepW[tid] = (tid == NMSW - 1) ? 0xFu : 0xFFFFFFFFu; // 900 = 28*32+4
  __syncthreads();

  // ---- Phase 4a: build suppression bitmask (j > i, iou > th) ----
  for (int i = tid; i < QN; i += 256) {
    float ax1 = s_box4[i][0], ay1 = s_box4[i][1];
    float ax2 = s_box4[i][2], ay2 = s_box4[i][3];
    float areaA = (ax2 - ax1) * (ay2 - ay1);
    int w0 = (i + 1) >> 5;
    for (int w = 0; w < NMSW; ++w) {
      unsigned bits = 0u;
      if (w >= w0) {
        int jb = w << 5;
        #pragma unroll 4
        for (int t = 0; t < 32; ++t) {
          int j = jb + t;
          if (j > i && j < QN) {
            float bx1 = s_box4[j][0], by1 = s_box4[j][1];
            float bx2 = s_box4[j][2], by2 = s_box4[j][3];
            float lt0 = fmaxf(ax1, bx1), lt1 = fmaxf(ay1, by1);
            float rb0 = fminf(ax2, bx2), rb1 = fminf(ay2, by2);
            float ww = fmaxf(rb0 - lt0, 0.f), hh = fmaxf(rb1 - lt1, 0.f);
            float inter = ww * hh;
            float areaB = (bx2 - bx1) * (by2 - by1);
            float un = areaA + areaB - inter;
            float iou = inter / ((un == 0.f) ? 1.f : un);
            if (iou > NMS_THF) bits |= (1u << t);
          }
        }
      }
      s_mask[i * NMSW + w] = bits;
    }
  }
  __syncthreads();

  // ---- Phase 4b: greedy resolve, single wave, no block barriers ----
  if (tid < 32) {
    for (int i = 0; i < QN; ++i) {
      unsigned kw = s_keepW[i >> 5];            // uniform read across wave
      if ((kw >> (i & 31)) & 1u) {
        if (tid < NMSW) s_keepW[tid] &= ~s_mask[i * NMSW + tid];
      }
    }
  }
  __syncthreads();

  // ---- Phase 5: scatter keep back to original order ----
  for (int r = tid; r < QN; r += 256)
    s_keepO[s_idx[r]] = (s_keepW[r >> 5] >> (r & 31)) & 1u;
  __syncthreads();

  // ---- Phase 6: top-k selection with exact top_k tie semantics ----
  if (tid == 0) {
    int nh = 0, no = 0;
    for (int q = 0; q < QN; ++q) {
      if (s_keepO[q] && s_score[q] >= SCORE_THF) {
        if (s_label[q] == 0) nh++; else no++;
      }
    }
    int kh = min(max(nh, 3), 15);
    int ko = min(max(no, 3), 15);
    int ch = 0, co = 0;
    for (int r = 0; r < QN && (ch < MAXH || co < MAXO); ++r) {
      int q = s_idx[r];
      if (s_keepO[q]) {
        if (s_label[q] == 0) {
          if (ch < MAXH) { s_idxh[ch] = q; s_sch[ch] = s_score[q]; ch++; }
        } else {
          if (co < MAXO) { s_idxo[co] = q; s_sco[co] = s_score[q]; co++; }
        }
      }
    }
    // NEG-filled slots: top_k is stable -> ascending original index among equals
    for (int q = 0; q < QN && ch < MAXH; ++q)
      if (!(s_keepO[q] && s_label[q] == 0)) { s_idxh[ch] = q; s_sch[ch] = NEGV; ch++; }
    for (int q = 0; q < QN && co < MAXO; ++q)
      if (!(s_keepO[q] && s_label[q] != 0)) { s_idxo[co] = q; s_sco[co] = NEGV; co++; }
    for (int t = 0; t < MAXH; ++t) {
      s_valh[t] = (t < kh) && (s_sch[t] > NEGV * 0.5f);
      s_valo[t] = (t < ko) && (s_sco[t] > NEGV * 0.5f);
    }
    s_idxh[MAXH] = 0;   // pad slot used by GEMM rows p >= 225
  }
  __syncthreads();

  if (tid < NPAIR) {
    int h = tid / 15, o = tid % 15;
    s_pr[tid] = (s_valh[h] && s_valo[o]) ? s_sch[h] * s_sco[o] : 0.f;
  }
  __syncthreads();

  // ---- Phase 7: pair GEMM with V_WMMA_F32_16X16X4_F32 (branch-free inner loop) ----
  // A = feat (225x512, rows padded to 240 via pad slot), B = Wc (512x128 packed)
  {
    const int wave = tid >> 5;
    const int lane = tid & 31;
    const int half = lane >> 4;   // 0: K {0,1} / C rows 0..7 ; 1: K {2,3} / C rows 8..15
    const int ln   = lane & 15;
    for (int t = wave; t < 120; t += 8) {
      int mtile = t >> 3, ntile = t & 7;
      int p = mtile * 16 + ln;                       // A row (pair index) for this lane
      const float* rowH = hb + (size_t)s_idxh[p / 15] * DN;   // p/15 <= 15 (pad slot)
      const float* rowO = hb + (size_t)s_idxo[p % 15] * DN;
      int n = ntile * 16 + ln;                       // B/C column
      const float* wcol = Wc + n;                    // column base, row stride NPAD
      v8f acc0 = {}, acc1 = {};
      #pragma unroll 4
      for (int kk = 0; kk < DN; kk += 4) {
        int c0 = kk + 2 * half;
        v2f aH = *(const v2f*)(rowH + c0);
        v2f aO = *(const v2f*)(rowO + c0);
        v2f bH; bH.x = wcol[c0 * NPAD];        bH.y = wcol[(c0 + 1) * NPAD];
        v2f bO; bO.x = wcol[(DN + c0) * NPAD]; bO.y = wcol[(DN + c0 + 1) * NPAD];
        acc0 = __builtin_amdgcn_wmma_f32_16x16x4_f32(
            false, aH, false, bH, (short)0, acc0, false, false);
        acc1 = __builtin_amdgcn_wmma_f32_16x16x4_f32(
            false, aO, false, bO, (short)0, acc1, false, false);
      }
      float bias = biasc[n];
      #pragma unroll
      for (int r = 0; r < 8; ++r) {
        int p2 = mtile * 16 + half * 8 + r;          // C row for VGPR r
        if (p2 >= NPAIR) continue;
        float val = acc0[r] + acc1[r] + bias;
        if (n < VN) {
          float sg = 1.f / (1.f + expf(-val));
          out_det[((size_t)b * NPAIR + p2) * VN + n] = sg * s_pr[p2];
        } else if (n < NTOT) {
          s_delta[p2][n - VN] = val;
        }
      }
    }
  }
  __syncthreads();

  // ---- Phase 8: box decode + outputs ----
  {
    float hgt = sizes[b * 2 + 0], wid = sizes[b * 2 + 1];
    if (tid < NPAIR) {
      int p = tid;
      int hI = s_idxh[p / 15], oI = s_idxo[p % 15];
      // human box
      {
        float cx = bx[hI * 4 + 0], cy = bx[hI * 4 + 1];
        float w  = bx[hI * 4 + 2], h  = bx[hI * 4 + 3];
        float d0 = s_delta[p][0], d1 = s_delta[p][1];
        float d2 = s_delta[p][2], d3 = s_delta[p][3];
        float cx2 = cx + d0 * w, cy2 = cy + d1 * h;
        float w2 = w * expf(d2), h2 = h * expf(d3);
        float* o = out_bh + ((size_t)b * NPAIR + p) * 4;
        o[0] = (cx2 - 0.5f * w2) * wid;
        o[1] = (cy2 - 0.5f * h2) * hgt;
        o[2] = (cx2 + 0.5f * w2) * wid;
        o[3] = (cy2 + 0.5f * h2) * hgt;
      }
      // object box
      {
        float cx = bx[oI * 4 + 0], cy = bx[oI * 4 + 1];
        float w  = bx[oI * 4 + 2], h  = bx[oI * 4 + 3];
        float d0 = s_delta[p][4], d1 = s_delta[p][5];
        float d2 = s_delta[p][6], d3 = s_delta[p][7];
        float cx2 = cx + d0 * w, cy2 = cy + d1 * h;
        float w2 = w * expf(d2), h2 = h * expf(d3);
        float* o = out_bo + ((size_t)b * NPAIR + p) * 4;
        o[0] = (cx2 - 0.5f * w2) * wid;
        o[1] = (cy2 - 0.5f * h2) * hgt;
        o[2] = (cx2 + 0.5f * w2) * wid;
        o[3] = (cy2 + 0.5f * h2) * hgt;
      }
    }
    if (tid < 30) {
      int q = (tid < 15) ? s_idxh[tid] : s_idxo[tid - 15];
      float cx = bx[q * 4 + 0], cy = bx[q * 4 + 1];
      float w  = bx[q * 4 + 2], h  = bx[q * 4 + 3];
      float* o = out_boxes + ((size_t)b * 30 + tid) * 4;
      o[0] = (cx - 0.5f * w) * wid;
      o[1] = (cy - 0.5f * h) * hgt;
      o[2] = (cx + 0.5f * w) * wid;
      o[3] = (cy + 0.5f * h) * hgt;
    }
  }
}

extern "C" void kernel_launch(void* const* d_in, const int* in_sizes, int n_in,
                              void* d_out, int out_size, void* d_ws, size_t ws_size,
                              hipStream_t stream) {
  (void)n_in; (void)out_size; (void)ws_size;
  const float* logits = (const float*)d_in[0];
  const float* boxes  = (const float*)d_in[1];
  const float* hsv    = (const float*)d_in[2];
  const float* sizes  = (const float*)d_in[3];
  const float* Wv     = (const float*)d_in[4];
  const float* bv     = (const float*)d_in[5];
  const float* Wb     = (const float*)d_in[6];
  const float* bb     = (const float*)d_in[7];

  const int B = in_sizes[3] / 2;   // image_sizes is (B, 2)

  float* Wc    = (float*)d_ws;            // 512*128 floats
  float* biasc = Wc + 2 * DN * NPAD;      // 128 floats

  float* out = (float*)d_out;
  float* out_boxes = out;                                   // B*30*4
  float* out_bh    = out_boxes + (size_t)B * 30 * 4;        // B*225*4
  float* out_bo    = out_bh    + (size_t)B * NPAIR * 4;     // B*225*4
  float* out_det   = out_bo    + (size_t)B * NPAIR * 4;     // B*225*117

  prep_w_kernel<<<(2 * DN * NPAD + 255) / 256, 256, 0, stream>>>(Wv, bv, Wb, bb, Wc, biasc);

  const size_t mask_bytes = (size_t)QN * NMSW * sizeof(unsigned);
  hoi_kernel<<<B, 256, mask_bytes, stream>>>(logits, boxes, hsv, sizes, Wc, biasc,
                                             out_boxes, out_bh, out_bo, out_det);
}